// DestroyEdgewise_20598663151962
// MI455X (gfx1250) — compile-verified
//
#include <hip/hip_runtime.h>
#include <hip/hip_bf16.h>

typedef __attribute__((ext_vector_type(2))) float v2f;
typedef __attribute__((ext_vector_type(8))) float v8f;

#define NEG_SLOPE 0.01f
#define KDIM 10
#define MDIM 64
#define HID 32
#define DD 64

__device__ __forceinline__ float leaky(float x) {
    return x > 0.f ? x : NEG_SLOPE * x;
}

// D = A(16x4 f32) * B(4x16 f32) + C(16x16 f32), full-precision WMMA
__device__ __forceinline__ v8f wmma_f32x4(v2f a, v2f b, v8f c) {
    return __builtin_amdgcn_wmma_f32_16x16x4_f32(
        /*neg_a=*/false, a, /*neg_b=*/false, b,
        /*c_mod=*/(short)0, c, /*reuse_a=*/false, /*reuse_b=*/false);
}

// ---------------------------------------------------------------- node encode
__global__ void node_encode_kernel(const float* __restrict__ coord,
                                   const float* __restrict__ Wn,
                                   const float* __restrict__ bn,
                                   float* __restrict__ nf, int nTot) {
    int gid = blockIdx.x * blockDim.x + threadIdx.x;
    if (gid >= nTot) return;
    int d = gid & 63;
    int n = gid >> 6;
    nf[gid] = coord[2 * n] * Wn[d] + coord[2 * n + 1] * Wn[64 + d] + bn[d];
}

// ---------------------------------------------------------------- zero scratch
__global__ void zero_kernel(float* __restrict__ p, int n) {
    int i = blockIdx.x * blockDim.x + threadIdx.x;
    if (i < n) p[i] = 0.f;
}

// ---------------------------------------------------------------- scatter-add
// One wave per edge; lanes cover the 64-dim feature row in two coalesced slices.
__global__ __launch_bounds__(256) void scatter_kernel(
    const float* __restrict__ nf, const int* __restrict__ esrc,
    const int* __restrict__ edst, float* __restrict__ agg, int nEdges) {
    int wave = threadIdx.x >> 5;
    int lane = threadIdx.x & 31;
    int e = blockIdx.x * 8 + wave;
    if (e >= nEdges) return;
    int s = esrc[e];
    int d = edst[e];
    float v0 = nf[(size_t)s * 64 + lane];
    float v1 = nf[(size_t)s * 64 + 32 + lane];
    unsafeAtomicAdd(&agg[(size_t)d * 64 + lane], v0);
    unsafeAtomicAdd(&agg[(size_t)d * 64 + 32 + lane], v1);
}

// ---------------------------------------------------------------- GNN layer
// nf[16 rows] @ Wself + agg[16 rows] @ Wnbr, bias + LeakyReLU + residual.
// One wave per 16-row block; 4 accumulated 16x16 N-tiles; K swept in steps of 4.
__global__ __launch_bounds__(256) void gnn_update_kernel(
    float* __restrict__ nf, const float* __restrict__ agg,
    const float* __restrict__ Wself, const float* __restrict__ Wnbr,
    const float* __restrict__ bg, int nRowBlocks) {
    const int wave = threadIdx.x >> 5;
    const int lane = threadIdx.x & 31;
    const int rb = blockIdx.x * 8 + wave;
    if (rb >= nRowBlocks) return;          // uniform per wave: EXEC stays all-1
    const int row0 = rb * 16;
    const int r = lane & 15;               // A-row / B,C-column
    const int h = lane >> 4;               // half-wave selector

    v8f acc[4];
    const v8f z8 = {0.f, 0.f, 0.f, 0.f, 0.f, 0.f, 0.f, 0.f};
#pragma unroll
    for (int nt = 0; nt < 4; ++nt) acc[nt] = z8;

    const float* aS = nf + (size_t)(row0 + r) * 64 + 2 * h;
    const float* aN = agg + (size_t)(row0 + r) * 64 + 2 * h;

    for (int k = 0; k < 64; k += 4) {
        v2f a_s = *(const v2f*)(aS + k);   // A[r][k+2h], A[r][k+2h+1]
        v2f a_n = *(const v2f*)(aN + k);
        const int krow = k + 2 * h;
#pragma unroll
        for (int nt = 0; nt < 4; ++nt) {
            const float* bs = Wself + (size_t)krow * 64 + nt * 16 + r;
            v2f b1v;
            b1v.x = bs[0];
            b1v.y = bs[64];                // W[krow+1][col]
            acc[nt] = wmma_f32x4(a_s, b1v, acc[nt]);
            const float* bn2 = Wnbr + (size_t)krow * 64 + nt * 16 + r;
            v2f b2v;
            b2v.x = bn2[0];
            b2v.y = bn2[64];
            acc[nt] = wmma_f32x4(a_n, b2v, acc[nt]);
        }
    }

#pragma unroll
    for (int nt = 0; nt < 4; ++nt) {
#pragma unroll
        for (int v = 0; v < 8; ++v) {
            int rr = v + 8 * h;
            int col = nt * 16 + r;
            float act = leaky(acc[nt][v] + bg[col]);
            nf[(size_t)(row0 + rr) * 64 + col] += act;   // residual
        }
    }
}

// ---------------------------------------------------------------- masked edges
// Only the B*K*M masked edges are ever consumed -> compute ef just for those.
// ef = nf[src] @ We[:64] + nf[dst] @ We[64:] + be, then MLP 64->32->1 per row.
__global__ __launch_bounds__(128) void edge_score_kernel(
    const float* __restrict__ nf, const int* __restrict__ esrc,
    const int* __restrict__ edst, const int* __restrict__ mask,
    const float* __restrict__ We, const float* __restrict__ be,
    const float* __restrict__ W1, const float* __restrict__ b1,
    const float* __restrict__ W2, const float* __restrict__ b2,
    float* __restrict__ scores) {
    __shared__ float sW1[DD * HID];
    __shared__ float sb1[HID];
    __shared__ float sW2[HID];
    __shared__ float sEf[4][DD * 16];      // transposed: [col*16 + rowInTile]

    for (int i = threadIdx.x; i < DD * HID; i += blockDim.x) sW1[i] = W1[i];
    if (threadIdx.x < HID) {
        sb1[threadIdx.x] = b1[threadIdx.x];
        sW2[threadIdx.x] = W2[threadIdx.x];
    }
    __syncthreads();

    const int wave = threadIdx.x >> 5;
    const int lane = threadIdx.x & 31;
    const int tile = blockIdx.x * 4 + wave;   // grid sized exactly: no tail
    const int r = lane & 15;
    const int h = lane >> 4;
    const int row = tile * 16 + r;

    const int e = mask[row];
    const int s = esrc[e];
    const int d = edst[e];

    v8f acc[4];
    const v8f z8 = {0.f, 0.f, 0.f, 0.f, 0.f, 0.f, 0.f, 0.f};
#pragma unroll
    for (int nt = 0; nt < 4; ++nt) acc[nt] = z8;

    const float* aS = nf + (size_t)s * 64 + 2 * h;
    const float* aD = nf + (size_t)d * 64 + 2 * h;
    const float* We2 = We + 64 * 64;

    for (int k = 0; k < 64; k += 4) {
        v2f a_s = *(const v2f*)(aS + k);
        v2f a_d = *(const v2f*)(aD + k);
        const int krow = k + 2 * h;
#pragma unroll
        for (int nt = 0; nt < 4; ++nt) {
            const float* bp = We + (size_t)krow * 64 + nt * 16 + r;
            v2f bv;
            bv.x = bp[0];
            bv.y = bp[64];
            acc[nt] = wmma_f32x4(a_s, bv, acc[nt]);
            const float* bq = We2 + (size_t)krow * 64 + nt * 16 + r;
            v2f bw;
            bw.x = bq[0];
            bw.y = bq[64];
            acc[nt] = wmma_f32x4(a_d, bw, acc[nt]);
        }
    }

    // ef tile -> LDS, transposed so the MLP reads are bank-conflict-free
#pragma unroll
    for (int nt = 0; nt < 4; ++nt) {
#pragma unroll
        for (int v = 0; v < 8; ++v) {
            int col = nt * 16 + r;
            int rr = v + 8 * h;
            sEf[wave][col * 16 + rr] = acc[nt][v] + be[col];
        }
    }
    __syncthreads();

    // per-row tiny MLP (lanes 0..15 each own one row of the tile)
    if (lane < 16) {
        float sc = 0.f;
        for (int j = 0; j < HID; ++j) {
            float sum = sb1[j];
            for (int dd = 0; dd < DD; ++dd)
                sum += sEf[wave][dd * 16 + r] * sW1[dd * HID + j];
            sc += leaky(sum) * sW2[j];
        }
        scores[row] = sc + b2[0];
    }
}

// ---------------------------------------------------------------- reduce+softmax
__global__ void reduce_softmax_kernel(const float* __restrict__ scores,
                                      float* __restrict__ out, int B) {
    int b = blockIdx.x * blockDim.x + threadIdx.x;
    if (b >= B) return;
    float x[KDIM];
    float mx = -3.0e38f;
#pragma unroll
    for (int k = 0; k < KDIM; ++k) {
        const float* p = scores + ((size_t)b * KDIM + k) * MDIM;
        float ssum = 0.f;
        for (int m = 0; m < MDIM; ++m) ssum += p[m];
        x[k] = ssum;
        mx = fmaxf(mx, ssum);
    }
    float den = 0.f;
#pragma unroll
    for (int k = 0; k < KDIM; ++k) {
        x[k] = expf(x[k] - mx);
        den += x[k];
    }
#pragma unroll
    for (int k = 0; k < KDIM; ++k) out[(size_t)b * KDIM + k] = x[k] / den;
}

// ---------------------------------------------------------------- launcher
extern "C" void kernel_launch(void* const* d_in, const int* in_sizes, int n_in,
                              void* d_out, int out_size, void* d_ws, size_t ws_size,
                              hipStream_t stream) {
    const float* coord = (const float*)d_in[0];
    const int* esrc    = (const int*)d_in[1];
    const int* edst    = (const int*)d_in[2];
    const int* mask    = (const int*)d_in[3];
    const float* Wn    = (const float*)d_in[4];
    const float* bn    = (const float*)d_in[5];
    const float* Wself = (const float*)d_in[6];
    const float* Wnbr  = (const float*)d_in[7];
    const float* bg    = (const float*)d_in[8];
    const float* We    = (const float*)d_in[9];
    const float* be    = (const float*)d_in[10];
    const float* W1    = (const float*)d_in[11];
    const float* b1    = (const float*)d_in[12];
    const float* W2    = (const float*)d_in[13];
    const float* b2    = (const float*)d_in[14];
    float* out = (float*)d_out;

    const int N = in_sizes[0] / 2;   // 50000 nodes
    const int E = in_sizes[1];       // 800000 edges
    const int R = in_sizes[3];       // B*K*M masked rows (20480)
    const int L = in_sizes[8] / 64;  // GNN layers (3)
    const int B = out_size / KDIM;   // batch graphs (32)

    float* nf     = (float*)d_ws;               // N*64 f32
    float* agg    = nf + (size_t)N * 64;        // N*64 f32
    float* scores = agg + (size_t)N * 64;       // R f32

    const int nfTot = N * 64;
    node_encode_kernel<<<(nfTot + 255) / 256, 256, 0, stream>>>(coord, Wn, bn, nf, nfTot);

    const int nRB = N / 16;  // N divisible by 16 (50000 = 3125*16)
    for (int l = 0; l < L; ++l) {
        zero_kernel<<<(nfTot + 255) / 256, 256, 0, stream>>>(agg, nfTot);
        scatter_kernel<<<(E + 7) / 8, 256, 0, stream>>>(nf, esrc, edst, agg, E);
        gnn_update_kernel<<<(nRB + 7) / 8, 256, 0, stream>>>(
            nf, agg, Wself + (size_t)l * 64 * 64, Wnbr + (size_t)l * 64 * 64,
            bg + (size_t)l * 64, nRB);
    }

    // R divisible by 64 (20480): 4 tiles of 16 rows per 128-thread block
    edge_score_kernel<<<R / 64, 128, 0, stream>>>(nf, esrc, edst, mask, We, be,
                                                  W1, b1, W2, b2, scores);

    reduce_softmax_kernel<<<(B + 63) / 64, 64, 0, stream>>>(scores, out, B);
}